// SelectiveSSM_56324201119970
// MI455X (gfx1250) — compile-verified
//
#include <hip/hip_runtime.h>
#include <hip/hip_bf16.h>

typedef __attribute__((ext_vector_type(2))) float v2f;
typedef __attribute__((ext_vector_type(4))) float v4f;
typedef __attribute__((ext_vector_type(8))) float v8f;

#define D_MODEL 1024
#define D_STATE 16
#define D_CONV  4
#define D_INNER 2048
#define BATCH   2
#define SEQ     1024
#define NTOK    (BATCH * SEQ)          // 2048
#define NPAD    48                     // x_proj N padded 33 -> 48

// ---------------------------------------------------------------------------
// fp32 WMMA GEMM:  C[M x N] = A[M x K] (row-major) * B[N x K]^T
// Wave tile = (MT*16) x (NT*16); block = 256 threads = WM x WN waves.
// One b128 load per lane feeds TWO v_wmma_f32_16x16x4_f32 (K-permutation of
// the accumulation: lo float pair -> WMMA #1, hi pair -> WMMA #2, applied
// identically to A and B fragments so the pairing stays consistent).
// Next K-chunk is prefetched into registers before the current chunk's WMMAs
// so the scheduler can use partial s_wait_loadcnt instead of draining.
// ---------------------------------------------------------------------------
template <int MT, int NT>
__device__ __forceinline__ void mma_chunk(const v4f (&af)[MT], const v4f (&bf)[NT],
                                          v8f (&acc)[MT][NT]) {
#pragma unroll
    for (int mt = 0; mt < MT; ++mt) {
        v2f alo = __builtin_shufflevector(af[mt], af[mt], 0, 1);
        v2f ahi = __builtin_shufflevector(af[mt], af[mt], 2, 3);
#pragma unroll
        for (int i = 0; i < NT; ++i) {
            v2f blo = __builtin_shufflevector(bf[i], bf[i], 0, 1);
            v2f bhi = __builtin_shufflevector(bf[i], bf[i], 2, 3);
            acc[mt][i] = __builtin_amdgcn_wmma_f32_16x16x4_f32(
                false, alo, false, blo, (short)0, acc[mt][i], false, false);
            acc[mt][i] = __builtin_amdgcn_wmma_f32_16x16x4_f32(
                false, ahi, false, bhi, (short)0, acc[mt][i], false, false);
        }
    }
}

template <int MT, int NT, int WM, int WN>
__global__ __launch_bounds__(256)
void gemm_f32_wmma(const float* __restrict__ A, const float* __restrict__ B,
                   float* __restrict__ C, int M, int N, int K) {
    const int wave  = threadIdx.x >> 5;
    const int lane  = threadIdx.x & 31;
    const int half  = lane >> 4;        // 0 | 1
    const int r     = lane & 15;
    const int waveM = wave % WM;
    const int waveN = wave / WM;
    const int m0 = (blockIdx.y * WM + waveM) * (MT * 16);
    const int n0 = (blockIdx.x * WN + waveN) * (NT * 16);

    const float* arow[MT];
#pragma unroll
    for (int mt = 0; mt < MT; ++mt)
        arow[mt] = A + (size_t)(m0 + mt * 16 + r) * K + 4 * half;
    const float* brow[NT];
#pragma unroll
    for (int i = 0; i < NT; ++i)
        brow[i] = B + (size_t)(n0 + i * 16 + r) * K + 4 * half;

    v8f acc[MT][NT];
#pragma unroll
    for (int mt = 0; mt < MT; ++mt)
#pragma unroll
        for (int i = 0; i < NT; ++i)
#pragma unroll
            for (int j = 0; j < 8; ++j) acc[mt][i][j] = 0.0f;

    // preload chunk 0 (8 global k per chunk)
    v4f af[MT], bf[NT];
#pragma unroll
    for (int mt = 0; mt < MT; ++mt) af[mt] = *(const v4f*)(arow[mt]);
#pragma unroll
    for (int i = 0; i < NT; ++i)    bf[i]  = *(const v4f*)(brow[i]);

    for (int k = 8; k < K; k += 8) {
        v4f an[MT], bn[NT];
#pragma unroll
        for (int mt = 0; mt < MT; ++mt) an[mt] = *(const v4f*)(arow[mt] + k);
#pragma unroll
        for (int i = 0; i < NT; ++i)    bn[i]  = *(const v4f*)(brow[i] + k);
        mma_chunk<MT, NT>(af, bf, acc);     // waits only on older loads
#pragma unroll
        for (int mt = 0; mt < MT; ++mt) af[mt] = an[mt];
#pragma unroll
        for (int i = 0; i < NT; ++i)    bf[i]  = bn[i];
    }
    mma_chunk<MT, NT>(af, bf, acc);

    // D layout: acc[mt][i][j] = C[m0 + mt*16 + j + 8*half][n0 + i*16 + r]
#pragma unroll
    for (int mt = 0; mt < MT; ++mt)
#pragma unroll
        for (int i = 0; i < NT; ++i) {
            float* cp = C + (size_t)(m0 + mt * 16 + 8 * half) * N + (n0 + i * 16 + r);
#pragma unroll
            for (int j = 0; j < 8; ++j) cp[(size_t)j * N] = acc[mt][i][j];
        }
}

// ---------------------------------------------------------------------------
// Depthwise causal conv1d (k=4) + bias + SiLU on the first half of XZ.
// ---------------------------------------------------------------------------
__global__ __launch_bounds__(256)
void conv_silu_kernel(const float* __restrict__ XZ, const float* __restrict__ cw,
                      const float* __restrict__ cb, float* __restrict__ U) {
    int idx = blockIdx.x * blockDim.x + threadIdx.x;   // (b,l,d), d fastest
    int d = idx % D_INNER;
    int l = (idx / D_INNER) % SEQ;
    int b = idx / (D_INNER * SEQ);
    float acc = cb[d];
#pragma unroll
    for (int j = 0; j < D_CONV; ++j) {
        int lj = l - (D_CONV - 1) + j;
        if (lj >= 0)
            acc += cw[d * D_CONV + j] *
                   XZ[((size_t)(b * SEQ + lj)) * (2 * D_INNER) + d];
    }
    U[idx] = acc / (1.0f + __expf(-acc));              // SiLU
}

// Zero-pad x_proj_w (33 x 2048) to (48 x 2048).
__global__ __launch_bounds__(256)
void pad_w_kernel(const float* __restrict__ W, float* __restrict__ Wp) {
    int idx = blockIdx.x * blockDim.x + threadIdx.x;   // 48*2048
    int r = idx / D_INNER;
    Wp[idx] = (r < 2 * D_STATE + 1) ? W[idx] : 0.0f;
}

// dt[b,l,d] = softplus(x_dbl[b,l,32] * dtw[d] + dtb[d])
__global__ __launch_bounds__(256)
void dt_kernel(const float* __restrict__ XD, const float* __restrict__ dtw,
               const float* __restrict__ dtb, float* __restrict__ DT) {
    int idx = blockIdx.x * blockDim.x + threadIdx.x;   // NTOK * D_INNER
    int d = idx % D_INNER;
    int t = idx / D_INNER;
    float v = XD[(size_t)t * NPAD + 2 * D_STATE] * dtw[d] + dtb[d];
    float sp = (v > 20.0f) ? v : __logf(1.0f + __expf(v));
    DT[idx] = sp;
}

// ---------------------------------------------------------------------------
// Selective scan: lane = (channel, state). 16-lane groups own a channel's 16
// states; y_t reduced with a shfl_xor butterfly. Next timestep's operands are
// prefetched so the serial chain pays ALU latency, not L2 latency.
// ---------------------------------------------------------------------------
__global__ __launch_bounds__(256)
void scan_kernel(const float* __restrict__ U, const float* __restrict__ XD,
                 const float* __restrict__ DT, const float* __restrict__ A_log,
                 const float* __restrict__ XZ, float* __restrict__ Y) {
    const int tid = threadIdx.x;
    const int s = tid & 15;                    // state index
    const int g = tid >> 4;                    // channel group in block (0..15)
    const int b = blockIdx.x / (D_INNER / 16); // batch
    const int d = (blockIdx.x % (D_INNER / 16)) * 16 + g;

    const float Acoef = -__expf(A_log[d * D_STATE + s]);
    float state = 0.0f;

    const float* xdrow = XD + (size_t)(b * SEQ) * NPAD;
    const float* dtp   = DT + (size_t)(b * SEQ) * D_INNER + d;
    const float* up    = U  + (size_t)(b * SEQ) * D_INNER + d;
    const float* zp    = XZ + (size_t)(b * SEQ) * (2 * D_INNER) + D_INNER + d;
    float*       yp    = Y  + (size_t)(b * SEQ) * D_INNER + d;

    float Bv  = xdrow[s];
    float Cv  = xdrow[D_STATE + s];
    float dtv = dtp[0];
    float uv  = up[0];

    for (int t = 0; t < SEQ; ++t) {
        // prefetch t+1 while the exp/fma/reduce chain of t runs
        float Bn = Bv, Cn = Cv, dtn = dtv, un = uv;
        if (t + 1 < SEQ) {
            const float* nrow = xdrow + (size_t)(t + 1) * NPAD;
            Bn  = nrow[s];
            Cn  = nrow[D_STATE + s];
            dtn = dtp[(size_t)(t + 1) * D_INNER];
            un  = up[(size_t)(t + 1) * D_INNER];
        }
        float abar = __expf(dtv * Acoef);
        state = fmaf(abar, state, dtv * uv * Bv);
        float y = state * Cv;
        y += __shfl_xor(y, 1, 16);
        y += __shfl_xor(y, 2, 16);
        y += __shfl_xor(y, 4, 16);
        y += __shfl_xor(y, 8, 16);
        if (s == 0) {
            float z = zp[(size_t)t * (2 * D_INNER)];
            float sz = z / (1.0f + __expf(-z));
            yp[(size_t)t * D_INNER] = y * sz;
        }
        Bv = Bn; Cv = Cn; dtv = dtn; uv = un;
    }
}

// ---------------------------------------------------------------------------
extern "C" void kernel_launch(void* const* d_in, const int* in_sizes, int n_in,
                              void* d_out, int out_size, void* d_ws, size_t ws_size,
                              hipStream_t stream) {
    (void)in_sizes; (void)n_in; (void)out_size; (void)ws_size;
    const float* x        = (const float*)d_in[0];  // (2,1024,1024)
    const float* in_w     = (const float*)d_in[1];  // (4096,1024)
    const float* conv_w   = (const float*)d_in[2];  // (2048,1,4)
    const float* conv_b   = (const float*)d_in[3];  // (2048)
    const float* A_log    = (const float*)d_in[4];  // (2048,16)
    const float* xproj_w  = (const float*)d_in[5];  // (33,2048)
    const float* dt_w     = (const float*)d_in[6];  // (2048,1)
    const float* dt_b     = (const float*)d_in[7];  // (2048)
    const float* out_w    = (const float*)d_in[8];  // (1024,2048)
    float* out = (float*)d_out;                     // (2,1024,1024)

    float* ws  = (float*)d_ws;
    float* XZ  = ws;                                   // 2048*4096
    float* U   = XZ + (size_t)NTOK * 2 * D_INNER;      // 2048*2048
    float* Wp  = U  + (size_t)NTOK * D_INNER;          // 48*2048
    float* XD  = Wp + (size_t)NPAD * D_INNER;          // 2048*48
    float* DT  = XD + (size_t)NTOK * NPAD;             // 2048*2048
    float* Y   = DT + (size_t)NTOK * D_INNER;          // 2048*2048

    // 1) in_proj: XZ[2048,4096] = X[2048,1024] * in_w[4096,1024]^T
    //    wave tile 32x64, block tile 128x128
    gemm_f32_wmma<2, 4, 4, 2><<<dim3((2 * D_INNER) / 128, NTOK / 128), 256, 0, stream>>>(
        x, in_w, XZ, NTOK, 2 * D_INNER, D_MODEL);

    // 2) depthwise conv + bias + SiLU -> U
    conv_silu_kernel<<<(NTOK * D_INNER) / 256, 256, 0, stream>>>(XZ, conv_w, conv_b, U);

    // 3) x_proj (N padded 33->48): XD[2048,48] = U * Wp^T ; block tile 256x48
    pad_w_kernel<<<(NPAD * D_INNER) / 256, 256, 0, stream>>>(xproj_w, Wp);
    gemm_f32_wmma<2, 3, 8, 1><<<dim3(1, NTOK / 256), 256, 0, stream>>>(
        U, Wp, XD, NTOK, NPAD, D_INNER);

    // 4) dt = softplus(x_dbl[...,32] * dt_w + dt_b)
    dt_kernel<<<(NTOK * D_INNER) / 256, 256, 0, stream>>>(XD, dt_w, dt_b, DT);

    // 5) selective scan + SiLU(z) gate -> Y
    scan_kernel<<<BATCH * (D_INNER / 16), 256, 0, stream>>>(U, XD, DT, A_log, XZ, Y);

    // 6) out_proj: out[2048,1024] = Y[2048,2048] * out_w[1024,2048]^T
    gemm_f32_wmma<2, 4, 4, 2><<<dim3(D_MODEL / 128, NTOK / 128), 256, 0, stream>>>(
        Y, out_w, out, NTOK, D_MODEL, D_INNER);
}